// ExtractionModel_28750511079887
// MI455X (gfx1250) — compile-verified
//
#include <hip/hip_runtime.h>
#include <math.h>

typedef __attribute__((ext_vector_type(2))) float v2f;
typedef __attribute__((ext_vector_type(8))) float v8f;

#define THRESH 0.5f
#define EPSN   1e-12f
#define H 32
#define W 32
#define C 128
#define PLANE (H*W)                 // 1024
#define P (C*PLANE)                 // 131072 keypoints per level
#define DESC_OFF  (3*P*2)           // 786432
#define SCORE_OFF (DESC_OFF + 3*P*C)// 51118080
#define MASK_OFF  (SCORE_OFF + 3*P) // 51511296

struct StencilOut {
    float w0, w1, w2, w3;           // tl, tr, bl, br bilinear weights
    int   o_tl, o_tr, o_bl, o_br;   // flat spatial offsets of corners
    float ki, kj, sm;               // refined coords, masked score
    int   mask;
};

__device__ __forceinline__ void stencil(const float* __restrict__ f,
                                        const float* __restrict__ s,
                                        int p, StencilOut& o) {
    int c = p >> 10, rem = p & (PLANE - 1);
    int i = rem >> 5, j = rem & (W - 1);
    const float* fc = f + c * PLANE;
    float fcc = fc[rem];
    float fm0 = (i > 0)     ? fc[rem - W] : 0.0f;
    float fp0 = (i < H - 1) ? fc[rem + W] : 0.0f;
    float f0m = (j > 0)     ? fc[rem - 1] : 0.0f;
    float f0p = (j < W - 1) ? fc[rem + 1] : 0.0f;
    float fmm = (i > 0     && j > 0)     ? fc[rem - W - 1] : 0.0f;
    float fmp = (i > 0     && j < W - 1) ? fc[rem - W + 1] : 0.0f;
    float fpm = (i < H - 1 && j > 0)     ? fc[rem + W - 1] : 0.0f;
    float fpp = (i < H - 1 && j < W - 1) ? fc[rem + W + 1] : 0.0f;
    float di  = 0.5f  * (fp0 - fm0);
    float dj  = 0.5f  * (f0p - f0m);
    float dii = fp0 - 2.0f * fcc + fm0;
    float djj = f0p - 2.0f * fcc + f0m;
    float dij = 0.25f * (fmm - fmp - fpm + fpp);
    float det = dii * djj - dij * dij;
    bool  det_ok = fabsf(det) > 1e-10f;
    float det_s  = det_ok ? det : 1.0f;
    float dispi = -( djj * di - dij * dj) / det_s;
    float dispj = -(-dij * di + dii * dj) / det_s;
    bool disp_ok = det_ok && (fabsf(dispi) < 0.5f) && (fabsf(dispj) < 0.5f);
    float sraw = s[p];
    float sth  = (sraw < THRESH) ? 0.0f : sraw;
    float ki = (disp_ok ? dispi : 0.0f) + (float)i;
    float kj = (disp_ok ? dispj : 0.0f) + (float)j;
    float i0f = floorf(ki), i1f = ceilf(ki);
    float j0f = floorf(kj), j1f = ceilf(kj);
    bool inb = (i0f >= 0.0f) && (i1f <= (float)(H - 1)) &&
               (j0f >= 0.0f) && (j1f <= (float)(W - 1));
    bool mask = (sth != 0.0f) && disp_ok && inb;
    int i0 = min(max((int)i0f, 0), H - 1), i1 = min(max((int)i1f, 0), H - 1);
    int j0 = min(max((int)j0f, 0), W - 1), j1 = min(max((int)j1f, 0), W - 1);
    float df = ki - i0f, ej = kj - j0f;
    o.w0 = (1.0f - df) * (1.0f - ej);
    o.w1 = (1.0f - df) * ej;
    o.w2 = df * (1.0f - ej);
    o.w3 = df * ej;
    o.o_tl = i0 * W + j0; o.o_tr = i0 * W + j1;
    o.o_bl = i1 * W + j0; o.o_br = i1 * W + j1;
    o.ki = ki; o.kj = kj;
    o.sm = mask ? sth : 0.0f;
    o.mask = mask ? 1 : 0;
}

// ---- Pass 1: deterministic per-block partial sums of masked s^2 -------------
__global__ void k_score_partial(const float* __restrict__ f0, const float* __restrict__ f1,
                                const float* __restrict__ f2, const float* __restrict__ s0,
                                const float* __restrict__ s1, const float* __restrict__ s2,
                                float* __restrict__ partial) {
    int g = blockIdx.x * 256 + threadIdx.x;     // 0 .. 3P-1
    int L = g >> 17;                            // P == 2^17
    int p = g & (P - 1);
    const float* f = (L == 0) ? f0 : ((L == 1) ? f1 : f2);
    const float* s = (L == 0) ? s0 : ((L == 1) ? s1 : s2);
    StencilOut o; stencil(f, s, p, o);
    __shared__ float red[256];
    red[threadIdx.x] = o.sm * o.sm;
    __syncthreads();
    for (int off = 128; off > 0; off >>= 1) {
        if (threadIdx.x < off) red[threadIdx.x] += red[threadIdx.x + off];
        __syncthreads();
    }
    if (threadIdx.x == 0) partial[blockIdx.x] = red[0];
}

// ---- Pass 2: per-level 1/max(sqrt(sum),eps) ---------------------------------
__global__ void k_score_norm(const float* __restrict__ partial, float* __restrict__ rnorm) {
    int L = blockIdx.x;                          // 3 blocks, 512 partials each
    float v = 0.0f;
    for (int k = threadIdx.x; k < 512; k += 256) v += partial[L * 512 + k];
    __shared__ float red[256];
    red[threadIdx.x] = v;
    __syncthreads();
    for (int off = 128; off > 0; off >>= 1) {
        if (threadIdx.x < off) red[threadIdx.x] += red[threadIdx.x + off];
        __syncthreads();
    }
    if (threadIdx.x == 0) rnorm[L] = 1.0f / fmaxf(sqrtf(red[0]), EPSN);
}

// ---- Pass 3: main kernel — wave per 16 keypoints, WMMA bilinear combine -----
__global__ void k_main(const float* __restrict__ f0, const float* __restrict__ f1,
                       const float* __restrict__ f2, const float* __restrict__ s0,
                       const float* __restrict__ s1, const float* __restrict__ s2,
                       const float* __restrict__ rnorm, float* __restrict__ out) {
    int lane  = threadIdx.x & 31;
    int wid   = threadIdx.x >> 5;
    int T     = blockIdx.x * 8 + wid;            // tile id, 0..24575 (exact grid)
    int L     = T >> 13;                         // 8192 tiles per level
    int pbase = (T & 8191) << 4;
    const float* f = (L == 0) ? f0 : ((L == 1) ? f1 : f2);
    const float* s = (L == 0) ? s0 : ((L == 1) ? s1 : s2);

    int n = lane & 15;                           // keypoint column / A-row index
    int p = pbase + n;
    StencilOut o; stencil(f, s, p, o);           // lanes 16-31 duplicate lanes 0-15

    bool hi = lane >= 16;
    const float* fl = f + n * PLANE;             // A row M = n; channel = t*16 + n

    // D[c,p] = sum_q A[c, 4q+k] * B[4q+k, p]; per K=4 step, only column q is
    // nonzero in B and carries keypoint q's four bilinear weights.
    v8f d[8] = {};                               // 8 channel-tiles of 16x16 f32 D
    for (int q = 0; q < 16; ++q) {
        int   otl = __shfl(o.o_tl, q, 32);
        int   otr = __shfl(o.o_tr, q, 32);
        int   obl = __shfl(o.o_bl, q, 32);
        int   obr = __shfl(o.o_br, q, 32);
        float w0q = __shfl(o.w0, q, 32);
        float w1q = __shfl(o.w1, q, 32);
        float w2q = __shfl(o.w2, q, 32);
        float w3q = __shfl(o.w3, q, 32);
        int ga0 = hi ? obl : otl;                // A VGPR0: K=0 (lo) / K=2 (hi)
        int ga1 = hi ? obr : otr;                // A VGPR1: K=1 (lo) / K=3 (hi)
        float b0 = (n == q) ? (hi ? w2q : w0q) : 0.0f;  // B rows K0/K2, col q
        float b1 = (n == q) ? (hi ? w3q : w1q) : 0.0f;  // B rows K1/K3, col q
        v2f bv = {b0, b1};
        #pragma unroll
        for (int t = 0; t < 8; ++t) {
            const float* fch = fl + t * 16 * PLANE;
            v2f av = {fch[ga0], fch[ga1]};
            d[t] = __builtin_amdgcn_wmma_f32_16x16x4_f32(
                false, av, false, bv, (short)0, d[t], false, false);
        }
    }

    // Per-keypoint sum of squares over 128 channels: this lane's 64 + mate's 64.
    float part = 0.0f;
    #pragma unroll
    for (int t = 0; t < 8; ++t)
        #pragma unroll
        for (int r = 0; r < 8; ++r) part += d[t][r] * d[t][r];
    float tot = part + __shfl_xor(part, 16, 32);
    float factor = o.mask ? (1.0f / fmaxf(sqrtf(tot), EPSN)) : 0.0f;

    // Descriptors: D VGPR r holds channel t*16 + r (+8 for hi lanes), column n.
    int g = L * P + p;
    float* descp = out + DESC_OFF + g * C + (hi ? 8 : 0);
    #pragma unroll
    for (int t = 0; t < 8; ++t) {
        v8f dt = d[t];
        float4 a = {dt[0] * factor, dt[1] * factor, dt[2] * factor, dt[3] * factor};
        float4 b = {dt[4] * factor, dt[5] * factor, dt[6] * factor, dt[7] * factor};
        *reinterpret_cast<float4*>(descp + t * 16)     = a;
        *reinterpret_cast<float4*>(descp + t * 16 + 4) = b;
    }

    if (!hi) {
        out[2 * g]     = o.mask ? (o.ki * 16.0f + 7.5f) : 0.0f;
        out[2 * g + 1] = o.mask ? (o.kj * 16.0f + 7.5f) : 0.0f;
        out[SCORE_OFF + g] = o.sm * rnorm[L];
        out[MASK_OFF + g]  = o.mask ? 1.0f : 0.0f;
    }
}

extern "C" void kernel_launch(void* const* d_in, const int* in_sizes, int n_in,
                              void* d_out, int out_size, void* d_ws, size_t ws_size,
                              hipStream_t stream) {
    const float* f0 = (const float*)d_in[0];
    const float* f1 = (const float*)d_in[1];
    const float* f2 = (const float*)d_in[2];
    const float* s0 = (const float*)d_in[3];
    const float* s1 = (const float*)d_in[4];
    const float* s2 = (const float*)d_in[5];
    float* out     = (float*)d_out;
    float* partial = (float*)d_ws;        // 1536 floats
    float* rnorm   = partial + 1536;      // 3 floats

    k_score_partial<<<1536, 256, 0, stream>>>(f0, f1, f2, s0, s1, s2, partial);
    k_score_norm  <<<3,    256, 0, stream>>>(partial, rnorm);
    k_main        <<<3072, 256, 0, stream>>>(f0, f1, f2, s0, s1, s2, rnorm, out);
}